// UniGATLayer_81905026334990
// MI455X (gfx1250) — compile-verified
//
#include <hip/hip_runtime.h>
#include <hip/hip_bf16.h>
#include <stdint.h>

#define C_DIM     128
#define N_EDGES_C 100000
#define CHUNK     64          // nnz entries per wave; sorted edge runs avg ~16

typedef float v2f __attribute__((ext_vector_type(2)));
typedef float v8f __attribute__((ext_vector_type(8)));

// Native fp32 global atomic add (global_atomic_add_f32, non-returning path).
__device__ __forceinline__ void atomic_add_f32(float* p, float v) {
    __hip_atomic_fetch_add(p, v, __ATOMIC_RELAXED, __HIP_MEMORY_SCOPE_AGENT);
}

// ---------------------------------------------------------------- zero fill
__global__ void zero_f4(float4* __restrict__ p, int n4) {
    int i = blockIdx.x * blockDim.x + threadIdx.x;
    if (i < n4) p[i] = make_float4(0.f, 0.f, 0.f, 0.f);
}

// ---------------------------------------------------- stage 1: edge scatter
// m01[edge_idx[e], :] += values[e] * x0[node_idx[e], :]
// One wave32 per CHUNK of nnz; lane owns 4 channels (float4). edge_idx is
// sorted, so we accumulate runs in registers and flush one atomic row per
// run (~16x fewer atomics than per-entry flushing). Run boundaries are
// wave-uniform -> no divergence.
__global__ void scatter_edges(const float* __restrict__ x0,
                              const int*   __restrict__ node_idx,
                              const int*   __restrict__ edge_idx,
                              const float* __restrict__ values,
                              float*       __restrict__ m01,
                              int nnz) {
    int wave  = threadIdx.x >> 5;
    int lane  = threadIdx.x & 31;
    int chunk = blockIdx.x * (blockDim.x >> 5) + wave;
    int e0 = chunk * CHUNK;
    if (e0 >= nnz) return;
    int e1 = min(e0 + CHUNK, nnz);

    float4 acc = make_float4(0.f, 0.f, 0.f, 0.f);
    int prev = edge_idx[e0];

    for (int e = e0; e < e1; ++e) {
        int edge = edge_idx[e];
        if (edge != prev) {                       // flush completed run
            float* dst = m01 + (size_t)prev * C_DIM + lane * 4;
            atomic_add_f32(dst + 0, acc.x);
            atomic_add_f32(dst + 1, acc.y);
            atomic_add_f32(dst + 2, acc.z);
            atomic_add_f32(dst + 3, acc.w);
            acc  = make_float4(0.f, 0.f, 0.f, 0.f);
            prev = edge;
        }
        int   node = node_idx[e];
        float v    = values[e];
        float4 x = ((const float4*)(x0 + (size_t)node * C_DIM))[lane];
        acc.x += v * x.x;
        acc.y += v * x.y;
        acc.z += v * x.z;
        acc.w += v * x.w;
    }
    float* dst = m01 + (size_t)prev * C_DIM + lane * 4;
    atomic_add_f32(dst + 0, acc.x);
    atomic_add_f32(dst + 1, acc.y);
    atomic_add_f32(dst + 2, acc.z);
    atomic_add_f32(dst + 3, acc.w);
}

// ------------------------------------------- stage 2: row dot (wave per row)
// out[row] = dot(X[row, 0:128], a[0:128]); wave32, float4 per lane.
__global__ void rowdot(const float* __restrict__ X,
                       const float* __restrict__ a,
                       float*       __restrict__ out,
                       int nrows) {
    int wave = threadIdx.x >> 5;
    int lane = threadIdx.x & 31;
    int row  = blockIdx.x * (blockDim.x >> 5) + wave;
    if (row >= nrows) return;
    const float4 x = ((const float4*)(X + (size_t)row * C_DIM))[lane];
    const float4 w = ((const float4*)a)[lane];
    float d = x.x * w.x + x.y * w.y + x.z * w.z + x.w * w.w;
#pragma unroll
    for (int off = 16; off > 0; off >>= 1) d += __shfl_xor(d, off, 32);
    if (lane == 0) out[row] = d;
}

// -------------------------------- stage 3: fp32 WMMA GEMM  C = A(Mx128) * B(128x128)
// 256 threads = 8 waves; block owns a 16(M) x 128(N) strip, wave w owns N-tile w.
// v_wmma_f32_16x16x4_f32 lane layout (ISA 7.12.2):
//   A 16x4: lanes 0-15 -> M=lane, {VGPR0,VGPR1} = K0,K1 ; lanes 16-31 -> K2,K3
//   B 4x16: lanes 0-15 -> N=lane, {VGPR0,VGPR1} = K0,K1 ; lanes 16-31 -> K2,K3
//   C/D 16x16: VGPR r = row r (lanes 0-15) / row r+8 (lanes 16-31), N = lane&15
__global__ void gemm_wmma_f32(const float* __restrict__ A,
                              const float* __restrict__ B,
                              float*       __restrict__ C,
                              int M) {
    int wave  = threadIdx.x >> 5;
    int lane  = threadIdx.x & 31;
    int mtile = blockIdx.x;
    int row   = mtile * 16 + (lane & 15);
    if (row >= M) return;
    int ncol  = wave * 16 + (lane & 15);
    int khalf = (lane >> 4) * 2;          // 0 for lanes 0-15, 2 for lanes 16-31

    const float* arow = A + (size_t)row * C_DIM;
    __builtin_prefetch(arow, 0, 3);       // global_prefetch_b8: seed A stream

    v8f acc = {};
#pragma unroll 4
    for (int k0 = 0; k0 < C_DIM; k0 += 4) {
        v2f a, b;
        const float* ap = arow + k0 + khalf;            // contiguous pair
        a.x = ap[0];
        a.y = ap[1];
        const float* bp = B + (size_t)(k0 + khalf) * C_DIM + ncol;
        b.x = bp[0];
        b.y = bp[C_DIM];
        acc = __builtin_amdgcn_wmma_f32_16x16x4_f32(
            /*neg_a=*/false, a, /*neg_b=*/false, b,
            /*c_mod=*/(short)0, acc, /*reuse_a=*/false, /*reuse_b=*/false);
    }

    int rbase = mtile * 16 + (lane >> 4) * 8;
#pragma unroll
    for (int r = 0; r < 8; ++r)
        C[(size_t)(rbase + r) * C_DIM + ncol] = acc[r];
}

// ------------------------------------------- stage 4: score + node scatter
// s = ELU(es[edge] + ns[node]) * values[e];  out[node,:] += s * xmsg[edge,:]
// One wave32 per CHUNK; lane owns 4 channels. xmsg row + es loaded once per
// sorted-edge run and reused (~16x less gather traffic).
__global__ void scatter_out(const float* __restrict__ xmsg,
                            const float* __restrict__ es,
                            const float* __restrict__ ns,
                            const int*   __restrict__ node_idx,
                            const int*   __restrict__ edge_idx,
                            const float* __restrict__ values,
                            float*       __restrict__ out,
                            int nnz) {
    int wave  = threadIdx.x >> 5;
    int lane  = threadIdx.x & 31;
    int chunk = blockIdx.x * (blockDim.x >> 5) + wave;
    int e0 = chunk * CHUNK;
    if (e0 >= nnz) return;
    int e1 = min(e0 + CHUNK, nnz);

    int    prev = -1;
    float4 msg  = make_float4(0.f, 0.f, 0.f, 0.f);
    float  se   = 0.f;

    for (int e = e0; e < e1; ++e) {
        int edge = edge_idx[e];
        if (edge != prev) {                        // new run: load row once
            msg  = ((const float4*)(xmsg + (size_t)edge * C_DIM))[lane];
            se   = es[edge];
            prev = edge;
        }
        int   node = node_idx[e];
        float s = se + ns[node];
        s = (s > 0.f) ? s : (__expf(s) - 1.f);     // ELU, alpha = 1
        s *= values[e];
        float* dst = out + (size_t)node * C_DIM + lane * 4;
        atomic_add_f32(dst + 0, s * msg.x);
        atomic_add_f32(dst + 1, s * msg.y);
        atomic_add_f32(dst + 2, s * msg.z);
        atomic_add_f32(dst + 3, s * msg.w);
    }
}

// ---------------------------------------------------------------- launcher
extern "C" void kernel_launch(void* const* d_in, const int* in_sizes, int n_in,
                              void* d_out, int out_size, void* d_ws, size_t ws_size,
                              hipStream_t stream) {
    (void)n_in; (void)ws_size;
    const float* x0       = (const float*)d_in[0];
    const int*   node_idx = (const int*)  d_in[1];
    const int*   edge_idx = (const int*)  d_in[2];
    const float* values   = (const float*)d_in[3];
    const float* weight   = (const float*)d_in[4];
    const float* attw     = (const float*)d_in[5];   // [0:128]=a_src, [128:256]=a_tgt

    const int nnz     = in_sizes[3];                 // values element count
    const int n_nodes = in_sizes[0] / C_DIM;
    const int n_edges = N_EDGES_C;

    float* out = (float*)d_out;
    (void)out_size;

    // Workspace layout (256B aligned slices)
    char*  ws  = (char*)d_ws;
    size_t off = 0;
    float* m01  = (float*)(ws + off); off += ((size_t)n_edges * C_DIM * sizeof(float) + 255) & ~(size_t)255;
    float* xmsg = (float*)(ws + off); off += ((size_t)n_edges * C_DIM * sizeof(float) + 255) & ~(size_t)255;
    float* es   = (float*)(ws + off); off += ((size_t)n_edges * sizeof(float) + 255) & ~(size_t)255;
    float* ns   = (float*)(ws + off);

    // 0) zero accumulators (must happen every call; harness poisons buffers)
    {
        int n4 = n_edges * C_DIM / 4;
        zero_f4<<<(n4 + 255) / 256, 256, 0, stream>>>((float4*)m01, n4);
        int o4 = n_nodes * C_DIM / 4;
        zero_f4<<<(o4 + 255) / 256, 256, 0, stream>>>((float4*)out, o4);
    }

    // chunked wave-per-run scatter config: 8 waves (256 thr) per block
    const int n_chunks = (nnz + CHUNK - 1) / CHUNK;
    const int sblocks  = (n_chunks + 7) / 8;

    // 1) gather + edge scatter-add (run-accumulated atomics)
    scatter_edges<<<sblocks, 256, 0, stream>>>(x0, node_idx, edge_idx, values,
                                               m01, nnz);

    // 2) attention dot products (wave32 per row)
    {
        int rows_per_block = 256 / 32;
        rowdot<<<(n_edges + rows_per_block - 1) / rows_per_block, 256, 0, stream>>>(
            m01, attw, es, n_edges);
        rowdot<<<(n_nodes + rows_per_block - 1) / rows_per_block, 256, 0, stream>>>(
            x0, attw + C_DIM, ns, n_nodes);
    }

    // 3) x_message = m01 @ weight via v_wmma_f32_16x16x4_f32
    gemm_wmma_f32<<<(n_edges + 15) / 16, 256, 0, stream>>>(m01, weight, xmsg, n_edges);

    // 4) ELU score + node scatter-add (xmsg row cached per sorted-edge run)
    scatter_out<<<sblocks, 256, 0, stream>>>(xmsg, es, ns, node_idx, edge_idx,
                                             values, out, nnz);
}